// GauntTensorProductFixedParity_27221502722610
// MI455X (gfx1250) — compile-verified
//
#include <hip/hip_runtime.h>
#include <hip/hip_bf16.h>

// Problem constants (from reference): N=131072, D=64, LMAX=4 -> S=25,
// grid RB=12 x RA=11 -> G=132. Pad G->144 (GP) and S->32 (SP) for 16x16 WMMA tiles.
#define N_TOT 131072
#define DD    64
#define SS    25
#define SP    32
#define GG    132
#define GP    144
#define GPS   145   // LDS row stride for h staging; 145 mod 64 = 17, coprime with banks
#define TM    64    // rows per block (4 waves x 16 rows)

typedef __attribute__((ext_vector_type(2))) float v2f;
typedef __attribute__((ext_vector_type(8))) float v8f;

// ---------------------------------------------------------------------------
// Stage 0: fold the linear layers into the SH->grid evaluation.
//   B1[d][g] = sum_s W1[d][s] * Y[g][s]   (zero-padded to GP columns)
//   B2[d][g] = sum_s W2[d][s] * Y[g][s]
//   Ywp[g][s] = Yw[g][s] zero-padded to [GP][SP]
// Tiny (~400K MACs) -- one small grid.
// ---------------------------------------------------------------------------
__global__ void precompute_kernel(const float* __restrict__ W1,
                                  const float* __restrict__ W2,
                                  const float* __restrict__ Y,
                                  const float* __restrict__ Yw,
                                  float* __restrict__ B1,
                                  float* __restrict__ B2,
                                  float* __restrict__ Ywp) {
    int tid    = blockIdx.x * blockDim.x + threadIdx.x;
    int stride = gridDim.x * blockDim.x;
    for (int i = tid; i < DD * GP; i += stride) {
        int d = i / GP, g = i % GP;
        float s1 = 0.f, s2 = 0.f;
        if (g < GG) {
            for (int s = 0; s < SS; ++s) {
                float y = Y[g * SS + s];
                s1 += W1[d * SS + s] * y;
                s2 += W2[d * SS + s] * y;
            }
        }
        B1[i] = s1;
        B2[i] = s2;
    }
    for (int i = tid; i < GP * SP; i += stride) {
        int g = i / SP, s = i % SP;
        Ywp[i] = (g < GG && s < SS) ? Yw[g * SS + s] : 0.f;
    }
}

// ---------------------------------------------------------------------------
// Stage 1 (fused main kernel), per block of TM=64 rows (4 wave32s):
//   g1 = X1*B1, g2 = X2*B2  via v_wmma_f32_16x16x4_f32   (full fp32 precision)
//   h  = g1 .* g2           (accumulator layouts match -> plain v8f multiply)
//   out = h * Ywp           via WMMA again, h transposed through LDS
// ---------------------------------------------------------------------------
__global__ void __launch_bounds__(128)
gaunt_fused_kernel(const float* __restrict__ X1,
                   const float* __restrict__ X2,
                   const float* __restrict__ B1,
                   const float* __restrict__ B2,
                   const float* __restrict__ Ywp,
                   float* __restrict__ out) {
    extern __shared__ float smem[];
    float* sB1 = smem;                 // [DD][GP]
    float* sB2 = sB1 + DD * GP;        // [DD][GP]
    float* sYw = sB2 + DD * GP;        // [GP][SP]
    float* sH  = sYw + GP * SP;        // [TM][GPS]

    const int tid  = threadIdx.x;
    const int lane = tid & 31;
    const int wave = tid >> 5;
    const int hsel = lane >> 4;        // 0 for lanes 0-15, 1 for 16-31
    const int l16  = lane & 15;

    // Cooperative load of the small constant matrices into LDS.
    for (int i = tid; i < DD * GP; i += blockDim.x) {
        sB1[i] = B1[i];
        sB2[i] = B2[i];
    }
    for (int i = tid; i < GP * SP; i += blockDim.x) sYw[i] = Ywp[i];
    __syncthreads();

    const int rowBase = blockIdx.x * TM + wave * 16;

    // Preload A operands for all 16 K-steps (K=64, 4 per WMMA).
    // fp32 16x4 A layout: lane = M (=l16), VGPR0 holds K, VGPR1 holds K+1,
    // and lanes 16-31 take K+2/K+3 -> contiguous float2 at column 4j+2*hsel.
    v2f A1[16], A2[16];
    {
        const float* x1p = X1 + (size_t)(rowBase + l16) * DD + 2 * hsel;
        const float* x2p = X2 + (size_t)(rowBase + l16) * DD + 2 * hsel;
#pragma unroll
        for (int j = 0; j < 16; ++j) {
            A1[j] = *(const v2f*)(x1p + 4 * j);
            A2[j] = *(const v2f*)(x2p + 4 * j);
        }
    }

    // GEMM1 + pointwise product, one 16-wide column tile of the grid at a time.
    for (int t = 0; t < GP / 16; ++t) {
        const int col = t * 16 + l16;
        v8f acc1 = {};
        v8f acc2 = {};
#pragma unroll
        for (int j = 0; j < 16; ++j) {
            const int k = 4 * j + 2 * hsel;   // B 4x16 layout mirrors A halves
            v2f b1, b2;
            b1.x = sB1[k * GP + col];
            b1.y = sB1[(k + 1) * GP + col];
            b2.x = sB2[k * GP + col];
            b2.y = sB2[(k + 1) * GP + col];
            acc1 = __builtin_amdgcn_wmma_f32_16x16x4_f32(
                false, A1[j], false, b1, (short)0, acc1, false, false);
            acc2 = __builtin_amdgcn_wmma_f32_16x16x4_f32(
                false, A2[j], false, b2, (short)0, acc2, false, false);
        }
        v8f h = acc1 * acc2;   // pointwise product on the sphere grid
        // D layout: VGPR r holds row r (lanes 0-15) / row r+8 (lanes 16-31).
#pragma unroll
        for (int r = 0; r < 8; ++r) {
            const int m = wave * 16 + r + 8 * hsel;
            sH[m * GPS + col] = h[r];
        }
    }
    // No barrier needed: each wave reads back only the sH rows it wrote
    // (same-wave LDS RAW handled by dscnt waits).

    // GEMM2: out[16 x SP] = h[16 x GP] * Ywp[GP x SP]; K = 144 -> 36 steps.
    for (int u = 0; u < SP / 16; ++u) {
        const int col = u * 16 + l16;
        const float* hrow = sH + (wave * 16 + l16) * GPS + 2 * hsel;
        v8f acc = {};
#pragma unroll
        for (int j = 0; j < GP / 4; ++j) {
            const int k = 4 * j;
            v2f a, b;
            a.x = hrow[k];
            a.y = hrow[k + 1];
            b.x = sYw[(k + 2 * hsel) * SP + col];
            b.y = sYw[(k + 2 * hsel + 1) * SP + col];
            acc = __builtin_amdgcn_wmma_f32_16x16x4_f32(
                false, a, false, b, (short)0, acc, false, false);
        }
        if (col < SS) {   // only cols 0..24 are real outputs
#pragma unroll
            for (int r = 0; r < 8; ++r) {
                const int m = r + 8 * hsel;
                out[(size_t)(rowBase + m) * SS + col] = acc[r];
            }
        }
    }
}

extern "C" void kernel_launch(void* const* d_in, const int* in_sizes, int n_in,
                              void* d_out, int out_size, void* d_ws, size_t ws_size,
                              hipStream_t stream) {
    const float* X1 = (const float*)d_in[0];
    const float* X2 = (const float*)d_in[1];
    const float* W1 = (const float*)d_in[2];
    const float* W2 = (const float*)d_in[3];
    const float* Y  = (const float*)d_in[4];
    const float* Yw = (const float*)d_in[5];

    float* B1  = (float*)d_ws;         // [DD][GP]
    float* B2  = B1 + DD * GP;         // [DD][GP]
    float* Ywp = B2 + DD * GP;         // [GP][SP]

    precompute_kernel<<<32, 256, 0, stream>>>(W1, W2, Y, Yw, B1, B2, Ywp);

    const size_t shmem =
        (size_t)(2 * DD * GP + GP * SP + TM * GPS) * sizeof(float);  // ~126 KB
    gaunt_fused_kernel<<<N_TOT / TM, 128, shmem, stream>>>(
        X1, X2, B1, B2, Ywp, (float*)d_out);
}